// CrossLaneInteraction_70291434766887
// MI455X (gfx1250) — compile-verified
//
#include <hip/hip_runtime.h>
#include <hip/hip_bf16.h>
#include <math.h>

#define NI    8192
#define NV    16384
#define NLAY  6
#define NCLS  3
#define EMBED 256
#define KDIM  265            // EMBED + 9
#define KPAD  268            // padded to multiple of 4
#define ASTRIDE 268          // LDS row stride (floats); conflict-free bank pattern
#define THRESH_LOGIT (-2.9444389791664403f)   // log(0.05/0.95)

typedef float v2f __attribute__((ext_vector_type(2)));
typedef float v8f __attribute__((ext_vector_type(8)));

// ---------------------------------------------------------------- calib ----
// calib = inv( veh2inf_rt[0]^T )  (double Gauss-Jordan, one thread)
__global__ void calib_kernel(const float* __restrict__ rt, float* __restrict__ calibF) {
    if (threadIdx.x != 0 || blockIdx.x != 0) return;
    double a[4][8];
    for (int i = 0; i < 4; ++i)
        for (int j = 0; j < 4; ++j) {
            a[i][j]     = (double)rt[j * 4 + i];   // transpose
            a[i][4 + j] = (i == j) ? 1.0 : 0.0;
        }
    for (int col = 0; col < 4; ++col) {
        int piv = col;
        for (int r = col + 1; r < 4; ++r)
            if (fabs(a[r][col]) > fabs(a[piv][col])) piv = r;
        if (piv != col)
            for (int j = 0; j < 8; ++j) { double t = a[col][j]; a[col][j] = a[piv][j]; a[piv][j] = t; }
        double inv = 1.0 / a[col][col];
        for (int j = 0; j < 8; ++j) a[col][j] *= inv;
        for (int r = 0; r < 4; ++r) if (r != col) {
            double f = a[r][col];
            for (int j = 0; j < 8; ++j) a[r][j] -= f * a[col][j];
        }
    }
    for (int i = 0; i < 4; ++i)
        for (int j = 0; j < 4; ++j)
            calibF[i * 4 + j] = (float)a[i][4 + j];
}

// ---------------------------------------------------------------- select ---
// Stable compaction of flat indices with sigmoid(cls) > 0.05 (== logit test),
// sel[k] = flat_index // 3, order ascending (matches jnp.nonzero).
__global__ __launch_bounds__(1024)
void select_kernel(const float* __restrict__ cls_last, int* __restrict__ sel) {
    __shared__ int scan[1024];
    __shared__ int base;
    const int tid = threadIdx.x;
    if (tid == 0) base = 0;
    __syncthreads();
    const int total = NI * NCLS;
    for (int start = 0; start < total; start += 1024) {
        int i = start + tid;
        int flag = (i < total && cls_last[i] > THRESH_LOGIT) ? 1 : 0;
        scan[tid] = flag;
        __syncthreads();
        for (int off = 1; off < 1024; off <<= 1) {
            int v = (tid >= off) ? scan[tid - off] : 0;
            __syncthreads();
            scan[tid] += v;
            __syncthreads();
        }
        if (flag) sel[base + scan[tid] - 1] = i / NCLS;
        __syncthreads();
        if (tid == 0) base += scan[1023];
        __syncthreads();
    }
}

// --------------------------------------------------------------- classes ---
__global__ void classes_kernel(const float* __restrict__ veh, const float* __restrict__ inf,
                               const int* __restrict__ sel, float* __restrict__ out, int Ntot) {
    int idx = blockIdx.x * blockDim.x + threadIdx.x;      // over NLAY*Ntot*NCLS
    if (idx >= NLAY * Ntot * NCLS) return;
    int l   = idx / (Ntot * NCLS);
    int rem = idx - l * Ntot * NCLS;
    int n   = rem / NCLS;
    int c   = rem - n * NCLS;
    float v;
    if (n < NV) v = veh[(l * NV + n) * NCLS + c];
    else        v = inf[(l * NI + sel[n - NV]) * NCLS + c];
    out[idx] = v;
}

// ---------------------------------------------------------------- coords ---
__global__ void coords_kernel(const float* __restrict__ veh, const float* __restrict__ inf,
                              const int* __restrict__ sel, const float* __restrict__ cb,
                              float* __restrict__ out, int Ntot) {
    int idx = blockIdx.x * blockDim.x + threadIdx.x;      // over NLAY*Ntot rows
    if (idx >= NLAY * Ntot) return;
    int l = idx / Ntot, n = idx - l * Ntot;
    float4 c;
    if (n < NV) {
        c = ((const float4*)veh)[l * NV + n];
    } else {
        int s = sel[n - NV];
        c = ((const float4*)inf)[l * NI + s];
        // norm2lidar (INF range) with z=0 -> Z = -5
        float X = c.x * 102.4f;
        float Y = c.y * 102.4f - 51.2f;
        float Z = -5.0f;
        float px = cb[0] * X + cb[1] * Y + cb[2]  * Z + cb[3];
        float py = cb[4] * X + cb[5] * Y + cb[6]  * Z + cb[7];
        c.x = (px + 51.2f) * (1.0f / 102.4f);
        c.y = (py + 51.2f) * (1.0f / 102.4f);
    }
    ((float4*)out)[idx] = c;
}

// ------------------------------------------------------------- reference ---
__device__ __forceinline__ float inv_sigmoid(float v) {
    float x  = fminf(fmaxf(v, 0.0f), 1.0f);
    float x1 = fmaxf(x, 1e-5f);
    float x2 = fmaxf(1.0f - x, 1e-5f);
    return __logf(x1 / x2);
}

__global__ void reference_kernel(const float* __restrict__ veh, const float* __restrict__ inf,
                                 const int* __restrict__ sel, const float* __restrict__ cb,
                                 float* __restrict__ out, int Ntot) {
    int n = blockIdx.x * blockDim.x + threadIdx.x;        // over Ntot rows
    if (n >= Ntot) return;
    float4 r;
    if (n < NV) {
        r = ((const float4*)veh)[n];
    } else {
        int s = sel[n - NV];
        r = ((const float4*)inf)[s];
        float sx = 1.0f / (1.0f + __expf(-r.x));
        float sy = 1.0f / (1.0f + __expf(-r.y));
        float X = sx * 102.4f;                 // norm2lidar INF, x: lo=0
        float Y = sy * 102.4f - 51.2f;
        float Z = 0.5f * 8.0f - 5.0f;          // sigmoid(0)=0.5 -> -1
        float px = cb[0] * X + cb[1] * Y + cb[2] * Z + cb[3];
        float py = cb[4] * X + cb[5] * Y + cb[6] * Z + cb[7];
        r.x = inv_sigmoid((px + 51.2f) * (1.0f / 102.4f));
        r.y = inv_sigmoid((py + 51.2f) * (1.0f / 102.4f));
    }
    ((float4*)out)[n] = r;
}

// --------------------------------------------------- WMMA align GEMM -------
// out[k, :] = concat(query[sel[k], :256], calib[:3,:3].flat) @ W + b
// One block (4 waves) per 16-row tile: A tile staged in LDS (16 x 268, zero
// padded), each wave computes 4 column tiles with V_WMMA_F32_16X16X4_F32.
// Main K loop is branch-free; the K tail loads W row 264 unconditionally and
// zeroes the out-of-range halves with lane selects (v_cndmask, no EXEC games).
__global__ __launch_bounds__(128)
void align_gemm_wmma(const float* __restrict__ q0, const float* __restrict__ q1,
                     const float* __restrict__ W0, const float* __restrict__ b0,
                     const float* __restrict__ W1, const float* __restrict__ b1,
                     const int* __restrict__ sel, const float* __restrict__ cb,
                     float* __restrict__ out0, float* __restrict__ out1, int Ksel) {
    const float* Q = (blockIdx.y == 0) ? q0  : q1;
    const float* W = (blockIdx.y == 0) ? W0  : W1;
    const float* B = (blockIdx.y == 0) ? b0  : b1;
    float*       O = (blockIdx.y == 0) ? out0 : out1;

    __shared__ float As[16 * ASTRIDE];
    __shared__ int   srow[16];

    const int tid     = threadIdx.x;
    const int rowBase = blockIdx.x * 16;

    // row indices (clamped duplicate for tail tiles)
    if (tid < 16) {
        int g = rowBase + tid;
        srow[tid] = sel[(g < Ksel) ? g : (Ksel - 1)];
    }
    __syncthreads();

    // stage A: 16 rows x 256 query features (coalesced), then 9 calib + 3 zeros
    for (int j = tid; j < 16 * EMBED; j += 128) {
        int rrow = j >> 8;
        int col  = j & (EMBED - 1);
        As[rrow * ASTRIDE + col] = Q[(size_t)srow[rrow] * EMBED + col];
    }
    if (tid < 16 * 12) {
        int rrow = tid / 12, c = tid - rrow * 12;
        As[rrow * ASTRIDE + EMBED + c] = (c < 9) ? cb[(c / 3) * 4 + (c % 3)] : 0.0f;
    }
    __syncthreads();

    const int lane  = tid & 31;
    const int wave  = tid >> 5;
    const int m     = lane & 15;       // A row / B,D column within tile
    const int khalf = lane >> 4;       // K-pair selector (ISA f32 16x4 layout)

    v8f acc[4];
#pragma unroll
    for (int t = 0; t < 4; ++t) acc[t] = (v8f){0.f,0.f,0.f,0.f,0.f,0.f,0.f,0.f};

    const float* ArowL = &As[m * ASTRIDE];   // LDS
    // ---- main loop: branch-free, all W rows < 265 ----
    for (int k0 = 0; k0 < 264; k0 += 4) {
        const int ka = k0 + khalf * 2;
        v2f av = *(const v2f*)(ArowL + ka);
#pragma unroll
        for (int t = 0; t < 4; ++t) {
            const int col = (wave * 4 + t) * 16 + m;
            v2f bv;
            bv.x = W[(size_t)ka * EMBED + col];
            bv.y = W[(size_t)(ka + 1) * EMBED + col];
            acc[t] = __builtin_amdgcn_wmma_f32_16x16x4_f32(
                         false, av, false, bv, (short)0, acc[t], false, false);
        }
    }
    // ---- tail step k0 = 264: A rows 265..267 are zero (LDS pre-padded);
    //      B row 264 loaded by all lanes, out-of-range halves zeroed by select.
    {
        const int ka = 264 + khalf * 2;
        v2f av = *(const v2f*)(ArowL + ka);
#pragma unroll
        for (int t = 0; t < 4; ++t) {
            const int col = (wave * 4 + t) * 16 + m;
            const float w264 = W[(size_t)264 * EMBED + col];   // unconditional
            v2f bv;
            bv.x = (khalf == 0) ? w264 : 0.0f;                  // v_cndmask
            bv.y = 0.0f;
            acc[t] = __builtin_amdgcn_wmma_f32_16x16x4_f32(
                         false, av, false, bv, (short)0, acc[t], false, false);
        }
    }

    // ---- store with bias: D layout lane(m,khalf): VGPR v -> row v + 8*khalf ----
#pragma unroll
    for (int t = 0; t < 4; ++t) {
        const int col  = (wave * 4 + t) * 16 + m;
        const float bb = B[col];
#pragma unroll
        for (int v = 0; v < 8; ++v) {
            int rr = rowBase + v + 8 * khalf;
            if (rr < Ksel)
                O[(size_t)rr * EMBED + col] = acc[t][v] + bb;
        }
    }
}

// ---------------------------------------------------------------------------
extern "C" void kernel_launch(void* const* d_in, const int* in_sizes, int n_in,
                              void* d_out, int out_size, void* d_ws, size_t ws_size,
                              hipStream_t stream) {
    const float* inf_cls   = (const float*)d_in[0];
    const float* inf_crd   = (const float*)d_in[1];
    const float* inf_q     = (const float*)d_in[2];
    const float* inf_qp    = (const float*)d_in[3];
    const float* inf_ref   = (const float*)d_in[4];
    const float* veh_cls   = (const float*)d_in[5];
    const float* veh_crd   = (const float*)d_in[6];
    const float* veh_q     = (const float*)d_in[7];
    const float* veh_qp    = (const float*)d_in[8];
    const float* veh_ref   = (const float*)d_in[9];
    const float* rt        = (const float*)d_in[10];
    const float* W_align   = (const float*)d_in[11];
    const float* b_align   = (const float*)d_in[12];
    const float* W_alignp  = (const float*)d_in[13];
    const float* b_alignp  = (const float*)d_in[14];
    float* out = (float*)d_out;

    // out_size = Ntot * (18 + 24 + 256 + 256 + 4) = Ntot * 558
    const int Ntot = out_size / 558;
    const int Ksel = Ntot - NV;

    // workspace: [0..64) calib 16 floats, [128..) sel[Ksel] ints
    float* calibF = (float*)d_ws;
    int*   sel    = (int*)((char*)d_ws + 128);

    calib_kernel<<<1, 32, 0, stream>>>(rt, calibF);
    if (Ksel > 0) {
        const float* cls_last = inf_cls + (size_t)(NLAY - 1) * NI * NCLS;
        select_kernel<<<1, 1024, 0, stream>>>(cls_last, sel);
    }

    float* out_cls = out;
    float* out_crd = out + (size_t)18 * Ntot;
    float* out_q   = out + (size_t)42 * Ntot;
    float* out_qp  = out + (size_t)298 * Ntot;
    float* out_ref = out + (size_t)554 * Ntot;

    {
        int tot = NLAY * Ntot * NCLS;
        classes_kernel<<<(tot + 255) / 256, 256, 0, stream>>>(veh_cls, inf_cls, sel, out_cls, Ntot);
    }
    {
        int tot = NLAY * Ntot;
        coords_kernel<<<(tot + 255) / 256, 256, 0, stream>>>(veh_crd, inf_crd, sel, calibF, out_crd, Ntot);
    }
    reference_kernel<<<(Ntot + 255) / 256, 256, 0, stream>>>(veh_ref, inf_ref, sel, calibF, out_ref, Ntot);

    // veh query blocks: contiguous D2D copies
    hipMemcpyAsync(out_q,  veh_q,  (size_t)NV * EMBED * sizeof(float), hipMemcpyDeviceToDevice, stream);
    hipMemcpyAsync(out_qp, veh_qp, (size_t)NV * EMBED * sizeof(float), hipMemcpyDeviceToDevice, stream);

    if (Ksel > 0) {
        float* out_q_inf  = out_q  + (size_t)NV * EMBED;
        float* out_qp_inf = out_qp + (size_t)NV * EMBED;
        int rowTiles = (Ksel + 15) / 16;
        dim3 grid(rowTiles, 2, 1);               // one block per 16-row tile; y: q vs q_pos
        align_gemm_wmma<<<grid, 128, 0, stream>>>(inf_q, inf_qp,
                                                  W_align, b_align, W_alignp, b_alignp,
                                                  sel, calibF, out_q_inf, out_qp_inf, Ksel);
    }
}